// Model_21088289423417
// MI455X (gfx1250) — compile-verified
//
#include <hip/hip_runtime.h>
#include <stdint.h>

typedef float v2f  __attribute__((ext_vector_type(2)));
typedef float v8f  __attribute__((ext_vector_type(8)));
typedef uint32_t u32x4 __attribute__((ext_vector_type(4)));
typedef int  i32x4 __attribute__((ext_vector_type(4)));
typedef int  i32x8 __attribute__((ext_vector_type(8)));

// ---- feature gates -------------------------------------------------------
#if defined(__has_builtin)
#if __has_builtin(__builtin_amdgcn_tensor_load_to_lds) && __has_builtin(__builtin_amdgcn_s_wait_tensorcnt)
#define HAVE_TDM 1
#else
#define HAVE_TDM 0
#endif
#else
#define HAVE_TDM 0
#endif

static __device__ __forceinline__ uint32_t lds_off(const void* p) {
  // generic pointer into LDS aperture: low 32 bits are the LDS byte address
  return (uint32_t)(uintptr_t)p;
}

#if HAVE_TDM
// 2D TDM load: tensor (dim1 rows x dim0 elems) of f32, tile == whole tensor.
static __device__ __forceinline__ void tdm_load_2d(const void* gaddr, uint32_t ldsaddr,
                                                   uint32_t dim0, uint32_t dim1,
                                                   uint32_t stride0) {
  uint64_t ga = (uint64_t)(uintptr_t)gaddr;
  u32x4 g0;
  g0[0] = 1u;                                         // count=1, is_restore=0
  g0[1] = ldsaddr;                                    // lds_addr [63:32]
  g0[2] = (uint32_t)(ga & 0xFFFFFFFFu);               // global_addr [95:64]
  g0[3] = (uint32_t)((ga >> 32) & 0x01FFFFFFu)        // global_addr [120:96]
        | (2u << 30);                                 // type=2 ("image") [127:126]
  i32x8 g1;
  g1[0] = (int)(2u << 16);                            // data_size=2 (4B), mask=0
  g1[1] = (int)((dim0 & 0xFFFFu) << 16);              // tensor_dim0 low16 @63:48
  g1[2] = (int)(((dim0 >> 16) & 0xFFFFu) | ((dim1 & 0xFFFFu) << 16)); // dim0 hi / dim1 lo
  g1[3] = (int)(((dim1 >> 16) & 0xFFFFu) | ((dim0 & 0xFFFFu) << 16)); // dim1 hi / tile_dim0
  g1[4] = (int)(dim1 & 0xFFFFu);                      // tile_dim1 (tile_dim2=0)
  g1[5] = (int)stride0;                               // tensor_dim0_stride low32
  g1[6] = 0;                                          // stride0 hi16 / stride1 lo16
  g1[7] = 0;
  i32x4 z4 = {0, 0, 0, 0};
#if __clang_major__ >= 23
  i32x8 z8 = {0, 0, 0, 0, 0, 0, 0, 0};
  __builtin_amdgcn_tensor_load_to_lds(g0, g1, z4, z4, z8, 0);
#else
  __builtin_amdgcn_tensor_load_to_lds(g0, g1, z4, z4, 0);
#endif
}
#endif

// ---- kernel 1: per-batch GEMM1 -> GEMM2 -> bias/PReLU -> mean / last row --
__global__ __launch_bounds__(256) void gcn_kernel(
    const float* __restrict__ seq1, const float* __restrict__ adj,
    const float* __restrict__ W_fc, const float* __restrict__ b_gcn,
    const float* __restrict__ prelu_a,
    float* __restrict__ c_out, float* __restrict__ hmv_out) {
  extern __shared__ float smem[];
  float* sA   = smem;           // 8192 f32: seq1[b] 64x128 (later: sS[0..4095], sH[4096..8191])
  float* sW   = smem + 8192;    // 8192 f32: W_fc 64x128 row-major [h][i]
  float* sAdj = smem + 16384;   // 4096 f32: adj[b] 64x64

  const int b    = blockIdx.x;
  const int tid  = threadIdx.x;
  const int lane = tid & 31;
  const int wave = tid >> 5;

#if HAVE_TDM
  if (wave == 0) {
    tdm_load_2d(seq1 + (size_t)b * 8192, lds_off(sA), 128u, 64u, 128u);
    tdm_load_2d(W_fc,                    lds_off(sW), 128u, 64u, 128u);
    tdm_load_2d(adj  + (size_t)b * 4096, lds_off(sAdj), 64u, 64u, 64u);
    __builtin_amdgcn_s_wait_tensorcnt(0);
  }
#else
  {
    const float4* gA = (const float4*)(seq1 + (size_t)b * 8192);
    const float4* gW = (const float4*)W_fc;
    const float4* gJ = (const float4*)(adj + (size_t)b * 4096);
    float4* dA = (float4*)sA; float4* dW = (float4*)sW; float4* dJ = (float4*)sAdj;
    for (int i = tid; i < 2048; i += 256) { dA[i] = gA[i]; dW[i] = gW[i]; }
    for (int i = tid; i < 1024; i += 256) { dJ[i] = gJ[i]; }
  }
#endif
  __syncthreads();

  // tile assignment: wave -> mTile (row strip), two adjacent nTiles
  const int mTile = wave >> 1;                 // 0..3
  const int nB    = (wave & 1) * 2;            // 0 or 2
  const int mrow  = mTile * 16 + (lane & 15);  // A-frag row for this lane
  const int kh    = (lane >> 4) << 1;          // K sub-offset per half-wave
  const int col0  = nB * 16 + (lane & 15);
  const int col1  = col0 + 16;

  // ---- GEMM1: seq_fts = seq1[b] (64x128) x W_fc^T (128x64) ----
  v8f acc0 = {0.f,0.f,0.f,0.f,0.f,0.f,0.f,0.f};
  v8f acc1 = {0.f,0.f,0.f,0.f,0.f,0.f,0.f,0.f};
#pragma unroll 4
  for (int k = 0; k < 128; k += 4) {
    const int kk = k + kh;
    v2f a  = *(const v2f*)(sA + mrow * 128 + kk);
    v2f b0 = *(const v2f*)(sW + col0 * 128 + kk);  // B[kk,col] = W_fc[col,kk]
    v2f b1 = *(const v2f*)(sW + col1 * 128 + kk);
    acc0 = __builtin_amdgcn_wmma_f32_16x16x4_f32(false, a, false, b0, (short)0, acc0, false, false);
    acc1 = __builtin_amdgcn_wmma_f32_16x16x4_f32(false, a, false, b1, (short)0, acc1, false, false);
  }
  __syncthreads();                      // all waves done reading sA

  // write seq_fts (64x64) into sS = smem[0..4095] (overlays sA)
  float* sS = smem;
  const int rbase = mTile * 16 + ((lane >> 4) << 3);
#pragma unroll
  for (int r = 0; r < 8; ++r) {
    sS[(rbase + r) * 64 + col0] = acc0[r];
    sS[(rbase + r) * 64 + col1] = acc1[r];
  }
  __syncthreads();

  // ---- GEMM2: out = adj[b] (64x64) x seq_fts (64x64) ----
  v8f d0 = {0.f,0.f,0.f,0.f,0.f,0.f,0.f,0.f};
  v8f d1 = {0.f,0.f,0.f,0.f,0.f,0.f,0.f,0.f};
#pragma unroll 4
  for (int k = 0; k < 64; k += 4) {
    const int kk = k + kh;
    v2f a = *(const v2f*)(sAdj + mrow * 64 + kk);
    v2f b0, b1;
    b0.x = sS[kk * 64 + col0]; b0.y = sS[(kk + 1) * 64 + col0];
    b1.x = sS[kk * 64 + col1]; b1.y = sS[(kk + 1) * 64 + col1];
    d0 = __builtin_amdgcn_wmma_f32_16x16x4_f32(false, a, false, b0, (short)0, d0, false, false);
    d1 = __builtin_amdgcn_wmma_f32_16x16x4_f32(false, a, false, b1, (short)0, d1, false, false);
  }

  // bias + PReLU, write h1 into sH = smem[4096..8191] (disjoint from sS reads)
  float* sH = smem + 4096;
  const float alpha = prelu_a[0];
  const float bias0 = b_gcn[col0];
  const float bias1 = b_gcn[col1];
#pragma unroll
  for (int r = 0; r < 8; ++r) {
    float v0 = d0[r] + bias0; v0 = (v0 >= 0.f) ? v0 : alpha * v0;
    float v1 = d1[r] + bias1; v1 = (v1 >= 0.f) ? v1 : alpha * v1;
    sH[(rbase + r) * 64 + col0] = v0;
    sH[(rbase + r) * 64 + col1] = v1;
  }
  __syncthreads();

  // c[b] = mean over rows 0..62, h_mv[b] = row 63
  if (tid < 64) {
    float s = 0.f;
#pragma unroll
    for (int r = 0; r < 63; ++r) s += sH[r * 64 + tid];
    c_out[(size_t)b * 64 + tid]   = s * (1.0f / 63.0f);
    hmv_out[(size_t)b * 64 + tid] = sH[63 * 64 + tid];
  }
}

// ---- kernel 2: bilinear scores (aligned + shifted c) ---------------------
__global__ __launch_bounds__(256) void bil_kernel(
    const float* __restrict__ W_bil, const float* __restrict__ b_bil,
    const float* __restrict__ c, const float* __restrict__ hmv,
    float* __restrict__ logits, int B) {
  __shared__ float sW[4096];
  for (int i = threadIdx.x; i < 4096; i += 256) sW[i] = W_bil[i];
  __syncthreads();
  const int b = blockIdx.x * 256 + threadIdx.x;
  if (b >= B) return;

  const float* hb = hmv + (size_t)b * 64;
  float u[64];
#pragma unroll
  for (int j = 0; j < 64; ++j) u[j] = 0.f;
  for (int i = 0; i < 64; ++i) {
    const float hi = hb[i];
#pragma unroll
    for (int j = 0; j < 64; ++j) u[j] = fmaf(hi, sW[i * 64 + j], u[j]);
  }
  const float* cb = c + (size_t)b * 64;
  const int bp = (b == 0) ? (B - 2) : (b - 1);
  const float* cp = c + (size_t)bp * 64;
  float s0 = 0.f, s1 = 0.f;
#pragma unroll
  for (int j = 0; j < 64; ++j) {
    s0 = fmaf(u[j], cb[j], s0);
    s1 = fmaf(u[j], cp[j], s1);
  }
  const float bb = b_bil[0];
  logits[b]     = s0 + bb;
  logits[B + b] = s1 + bb;
}

extern "C" void kernel_launch(void* const* d_in, const int* in_sizes, int n_in,
                              void* d_out, int out_size, void* d_ws, size_t ws_size,
                              hipStream_t stream) {
  const float* seq1    = (const float*)d_in[0];
  const float* adj     = (const float*)d_in[1];
  const float* W_fc    = (const float*)d_in[2];
  const float* b_gcn   = (const float*)d_in[3];
  const float* prelu_a = (const float*)d_in[4];
  const float* W_bil   = (const float*)d_in[5];
  const float* b_bil   = (const float*)d_in[6];

  const int B = in_sizes[0] / (64 * 128);   // 8192

  // d_out layout: logits(2B) | h_mv(B*64) | c(B*64)
  float* logits = (float*)d_out;
  float* hmv    = logits + 2 * B;
  float* cbuf   = hmv + (size_t)B * 64;

  const size_t smem_bytes = 20480 * sizeof(float);  // 80 KB dynamic LDS
  gcn_kernel<<<B, 256, smem_bytes, stream>>>(seq1, adj, W_fc, b_gcn, prelu_a, cbuf, hmv);
  bil_kernel<<<(B + 255) / 256, 256, 0, stream>>>(W_bil, b_bil, cbuf, hmv, logits, B);
}